// AttentionLayer_49005576847925
// MI455X (gfx1250) — compile-verified
//
#include <hip/hip_runtime.h>
#include <hip/hip_bf16.h>

// ---------------------------------------------------------------------------
// Bahdanau additive attention, fused for MI455X (gfx1250, wave32).
//   B=4, Te=512, Td=256, D=256
//   out = [ context (4*256*256 f32) | attention_weights (4*256*512 f32) ]
// GEMMs use V_WMMA_F32_16X16X4_F32 (full f32, matches reference precision).
// The huge [B,Td,Te,D] tanh intermediate is never materialized: the V_a dot
// is fused into the tanh loop. That loop (134M tanh) is the bottleneck, so
// it uses the CDNA5 hardware V_TANH_F32 (TRANS pipe, co-executes with the
// FMA accumulation) instead of the ~15-op OCML software tanh.
// ---------------------------------------------------------------------------

typedef __attribute__((ext_vector_type(2))) float v2f;
typedef __attribute__((ext_vector_type(8))) float v8f;

#define LATENT 256
#define TE 512
#define TD 256
#define NB 4

__device__ __forceinline__ float fast_tanh(float x) {
#if __has_builtin(__builtin_amdgcn_tanhf)
    return __builtin_amdgcn_tanhf(x);        // v_tanh_f32
#else
    float r;
    asm("v_tanh_f32 %0, %1" : "=v"(r) : "v"(x));
    return r;
#endif
}

// Generic row-major f32 GEMM: C[M,N] = A[M,K] * B[K,N], optional batch strides.
// One wave computes one 16x16 C tile; K consumed 4 at a time via WMMA.
// block = (32,4): 4 waves -> 4 consecutive M tiles, same N tile.
// grid  = (N/16, M/64, batches)
__global__ __launch_bounds__(128)
void gemm_f32_wmma(const float* __restrict__ A, const float* __restrict__ B,
                   float* __restrict__ C, int M, int N, int K,
                   long sA, long sB, long sC) {
    const int lane = threadIdx.x;          // 0..31
    const int half = lane >> 4;            // 0: lanes 0-15, 1: lanes 16-31
    const int l    = lane & 15;

    const long bz = blockIdx.z;
    A += bz * sA;  B += bz * sB;  C += bz * sC;

    const int tileN = blockIdx.x * 16;
    const int tileM = (blockIdx.y * 4 + threadIdx.y) * 16;

    // A row this lane feeds (ISA 7.12.2: A 16x4, lanes 0-15 hold K0/K1,
    // lanes 16-31 hold K2/K3, row M = lane%16).
    const float* arow = A + (long)(tileM + l) * K;
    const int col = tileN + l;

    v8f acc = {};   // C = 0 accumulate

    #pragma unroll 4
    for (int k0 = 0; k0 < K; k0 += 4) {
        const int kk = k0 + 2 * half;
        v2f a, b;
        a.x = arow[kk];
        a.y = arow[kk + 1];
        // B 4x16: rows striped across lanes within a VGPR (row kk for this
        // half-wave), matching the C/D row-split convention.
        b.x = B[(long)kk * N + col];
        b.y = B[(long)(kk + 1) * N + col];
        acc = __builtin_amdgcn_wmma_f32_16x16x4_f32(
                  false, a, false, b, (short)0, acc, false, false);
    }

    // C/D layout: VGPR r -> (row tileM+r+8*half, col tileN + lane%16)
    #pragma unroll
    for (int r = 0; r < 8; ++r) {
        const int row = tileM + r + 8 * half;
        C[(long)row * N + col] = acc[r];
    }
}

// One block per (b,t): fused score (tanh + V-dot) and softmax over Te.
// grid = (Td, B), block = 256 threads.
__global__ __launch_bounds__(256)
void score_softmax(const float* __restrict__ Wenc,   // [B,Te,D]
                   const float* __restrict__ Udec,   // [B,Td,D]
                   const float* __restrict__ Va,     // [D]
                   float* __restrict__ attn) {       // [B,Td,Te]
    __shared__ float u[LATENT];
    __shared__ float v[LATENT];
    __shared__ float sc[TE];
    __shared__ float red[256];

    const int t   = blockIdx.x;
    const int b   = blockIdx.y;
    const int tid = threadIdx.x;

    u[tid] = Udec[((long)b * TD + t) * LATENT + tid];
    v[tid] = Va[tid];
    __syncthreads();

    // scores: thread tid handles e = tid and e = tid + 256
    #pragma unroll
    for (int e = tid; e < TE; e += 256) {
        const float4* w4 = (const float4*)(Wenc + ((long)b * TE + e) * LATENT);
        float s = 0.f;
        #pragma unroll 4
        for (int q = 0; q < LATENT / 4; ++q) {
            const float4 w = w4[q];
            const int d = 4 * q;
            s += v[d + 0] * fast_tanh(w.x + u[d + 0]);
            s += v[d + 1] * fast_tanh(w.y + u[d + 1]);
            s += v[d + 2] * fast_tanh(w.z + u[d + 2]);
            s += v[d + 3] * fast_tanh(w.w + u[d + 3]);
        }
        sc[e] = s;
    }
    __syncthreads();

    // softmax over Te=512
    red[tid] = fmaxf(sc[tid], sc[tid + 256]);
    __syncthreads();
    for (int off = 128; off > 0; off >>= 1) {
        if (tid < off) red[tid] = fmaxf(red[tid], red[tid + off]);
        __syncthreads();
    }
    const float mx = red[0];
    __syncthreads();

    const float e0 = __expf(sc[tid] - mx);
    const float e1 = __expf(sc[tid + 256] - mx);
    red[tid] = e0 + e1;
    __syncthreads();
    for (int off = 128; off > 0; off >>= 1) {
        if (tid < off) red[tid] += red[tid + off];
        __syncthreads();
    }
    const float inv = 1.f / red[0];

    float* arow = attn + ((long)b * TD + t) * TE;
    arow[tid]       = e0 * inv;
    arow[tid + 256] = e1 * inv;
}

extern "C" void kernel_launch(void* const* d_in, const int* in_sizes, int n_in,
                              void* d_out, int out_size, void* d_ws, size_t ws_size,
                              hipStream_t stream) {
    const float* enc = (const float*)d_in[0];   // [4,512,256]
    const float* dec = (const float*)d_in[1];   // [4,256,256]
    const float* Wa  = (const float*)d_in[2];   // [256,256]
    const float* Ua  = (const float*)d_in[3];   // [256,256]
    const float* Va  = (const float*)d_in[4];   // [256,1]

    float* ctx  = (float*)d_out;                          // [4,256,256]
    float* attn = (float*)d_out + (long)NB * TD * LATENT; // [4,256,512]

    float* Wenc = (float*)d_ws;                           // [4*512,256] = 2 MB
    float* Udec = Wenc + (long)NB * TE * LATENT;          // [4*256,256] = 1 MB

    const dim3 blk(32, 4, 1);

    // W_enc = enc(2048x256) @ W_a(256x256)
    gemm_f32_wmma<<<dim3(LATENT / 16, (NB * TE) / 64, 1), blk, 0, stream>>>(
        enc, Wa, Wenc, NB * TE, LATENT, LATENT, 0, 0, 0);

    // U_dec = dec(1024x256) @ U_a(256x256)
    gemm_f32_wmma<<<dim3(LATENT / 16, (NB * TD) / 64, 1), blk, 0, stream>>>(
        dec, Ua, Udec, NB * TD, LATENT, LATENT, 0, 0, 0);

    // fused tanh-score + softmax -> attention weights
    score_softmax<<<dim3(TD, NB, 1), 256, 0, stream>>>(Wenc, Udec, Va, attn);

    // context[b] = attn[b](256x512) @ enc[b](512x256), batched over b
    gemm_f32_wmma<<<dim3(LATENT / 16, TD / 64, NB), blk, 0, stream>>>(
        attn, enc, ctx, TD, LATENT, TE,
        (long)TD * TE, (long)TE * LATENT, (long)TD * LATENT);
}